// FF_83519934038262
// MI455X (gfx1250) — compile-verified
//
#include <hip/hip_runtime.h>
#include <math.h>

#define D_DIM   1024
#define M_DIM   256
#define B_TILE  16
#define CHUNK   64
#define NCHUNK  (D_DIM / CHUNK)

typedef __attribute__((ext_vector_type(2))) float v2f;
typedef __attribute__((ext_vector_type(8))) float v8f;

typedef __attribute__((address_space(1))) int gint;   // global int
typedef __attribute__((address_space(3))) int lint;   // LDS int

// ---------- CDNA5 async global->LDS copy + asynccnt wait ----------------
#if defined(__has_builtin)
#if __has_builtin(__builtin_amdgcn_global_load_async_to_lds_b32)
#define HAVE_ASYNC_BUILTIN 1
#endif
#if __has_builtin(__builtin_amdgcn_s_wait_asynccnt)
#define HAVE_WAITASYNC_BUILTIN 1
#endif
#endif

#ifdef HAVE_ASYNC_BUILTIN
#define ASYNC_COPY_B32(g, l)                                                 \
  __builtin_amdgcn_global_load_async_to_lds_b32((gint*)(g), (lint*)(l), 0, 0)
#else
#define ASYNC_COPY_B32(g, l)                                                 \
  asm volatile("global_load_async_to_lds_b32 %0, %1, off"                    \
               :: "v"((unsigned)(unsigned long long)(lint*)(l)),             \
                  "v"(g) : "memory")
#endif

#ifdef HAVE_WAITASYNC_BUILTIN
#define WAIT_ASYNC(n) __builtin_amdgcn_s_wait_asynccnt(n)
#else
#define WAIT_ASYNC(n) asm volatile("s_wait_asynccnt " #n ::: "memory")
#endif

// ---------- prep kernel 1: logw = log_softmax(W), M_DIM threads ----------
__global__ void __launch_bounds__(M_DIM) prep_logw(const float* __restrict__ W,
                                                   float* __restrict__ logw) {
  __shared__ float red[8];
  const int t = threadIdx.x;
  const float w = W[t];

  float m = w;
  #pragma unroll
  for (int off = 16; off; off >>= 1) m = fmaxf(m, __shfl_xor(m, off, 32));
  if ((t & 31) == 0) red[t >> 5] = m;
  __syncthreads();
  float bm = red[0];
  #pragma unroll
  for (int i = 1; i < 8; ++i) bm = fmaxf(bm, red[i]);
  __syncthreads();

  float s = __expf(w - bm);
  #pragma unroll
  for (int off = 16; off; off >>= 1) s += __shfl_xor(s, off, 32);
  if ((t & 31) == 0) red[t >> 5] = s;
  __syncthreads();
  float bs = 0.f;
  #pragma unroll
  for (int i = 0; i < 8; ++i) bs += red[i];

  logw[t] = w - bm - __logf(bs);
}

// ---------- prep kernel 2: ac[d][m] = {2*sigmoid(P)-1, 1-sigmoid(P)} -----
__global__ void __launch_bounds__(256) prep_ac(const float* __restrict__ P,
                                               float2* __restrict__ ac) {
  const int gi = blockIdx.x * 256 + threadIdx.x;   // gi = d*M + m
  if (gi >= D_DIM * M_DIM) return;
  const int d = gi / M_DIM;
  const int m = gi - d * M_DIM;
  const float v = P[(size_t)m * D_DIM + d];
  const float p = 1.0f / (1.0f + __expf(-v));
  ac[gi] = make_float2(fmaf(2.0f, p, -1.0f), 1.0f - p);  // lik = c + x*a
}

// ---------- main kernel: 16 batch rows per block, thread == component m --
__global__ void __launch_bounds__(M_DIM) mixll_kernel(
    const float* __restrict__ x, const float2* __restrict__ ac,
    const float* __restrict__ logw, float* __restrict__ out) {
  // x tile stored TRANSPOSED: [dd][row], double buffered (8 KB)
  __shared__ float xt[2 * CHUNK * B_TILE];
  __shared__ float cll[B_TILE * M_DIM];     // 16 KB
  __shared__ float rowmax[B_TILE];

  const int t  = threadIdx.x;               // == m
  const int b0 = blockIdx.x * B_TILE;

  // async-load mapping: lane handles column dd_l for 4 rows (coalesced gmem)
  const int dd_l = t & (CHUNK - 1);
  const int r4   = (t >> 6) << 2;           // 0,4,8,12

  float acc[B_TILE];
  v2f prod2[B_TILE / 2];                    // packed row pairs -> v_pk_*_f32
  #pragma unroll
  for (int r = 0; r < B_TILE; ++r) acc[r] = 0.f;
  #pragma unroll
  for (int i = 0; i < B_TILE / 2; ++i) prod2[i] = (v2f){1.f, 1.f};

  // issue chunk 0 into buffer 0
  #pragma unroll
  for (int rr = 0; rr < 4; ++rr) {
    const int row = r4 + rr;
    ASYNC_COPY_B32(x + (size_t)(b0 + row) * D_DIM + dd_l,
                   &xt[dd_l * B_TILE + row]);
  }

  for (int c = 0; c < NCHUNK; ++c) {
    const int buf = (c & 1) * (CHUNK * B_TILE);
    if (c + 1 < NCHUNK) {
      const int nbuf = ((c + 1) & 1) * (CHUNK * B_TILE);
      #pragma unroll
      for (int rr = 0; rr < 4; ++rr) {
        const int row = r4 + rr;
        ASYNC_COPY_B32(x + (size_t)(b0 + row) * D_DIM + (c + 1) * CHUNK + dd_l,
                       &xt[nbuf + dd_l * B_TILE + row]);
      }
      WAIT_ASYNC(4);      // 4 prefetches may stay outstanding; older 4 done
    } else {
      WAIT_ASYNC(0);
    }
    __syncthreads();

    const float2* acp = ac + (size_t)c * CHUNK * M_DIM + t;
    for (int dd0 = 0; dd0 < CHUNK; dd0 += 16) {
      #pragma unroll
      for (int dd = 0; dd < 16; ++dd) {
        const float2 av = acp[(size_t)(dd0 + dd) * M_DIM];   // one b64 load
        const v2f aa = {av.x, av.x};
        const v2f cc = {av.y, av.y};
        // broadcast b128 LDS reads -> 4 consecutive VGPRs (even-aligned pairs)
        const float4* xv = (const float4*)&xt[buf + (dd0 + dd) * B_TILE];
        const float4 xa = xv[0], xb = xv[1], xc = xv[2], xd = xv[3];
        const v2f x0 = {xa.x, xa.y}, x1 = {xa.z, xa.w};
        const v2f x2 = {xb.x, xb.y}, x3 = {xb.z, xb.w};
        const v2f x4 = {xc.x, xc.y}, x5 = {xc.z, xc.w};
        const v2f x6 = {xd.x, xd.y}, x7 = {xd.z, xd.w};
        prod2[0] *= __builtin_elementwise_fma(x0, aa, cc);  // v_pk_fma_f32
        prod2[1] *= __builtin_elementwise_fma(x1, aa, cc);  // + v_pk_mul_f32
        prod2[2] *= __builtin_elementwise_fma(x2, aa, cc);
        prod2[3] *= __builtin_elementwise_fma(x3, aa, cc);
        prod2[4] *= __builtin_elementwise_fma(x4, aa, cc);
        prod2[5] *= __builtin_elementwise_fma(x5, aa, cc);
        prod2[6] *= __builtin_elementwise_fma(x6, aa, cc);
        prod2[7] *= __builtin_elementwise_fma(x7, aa, cc);
      }
      // one log per 16 factors: products stay in f32 normal range
      #pragma unroll
      for (int i = 0; i < B_TILE / 2; ++i) {
        acc[2 * i]     += __log2f(prod2[i].x);
        acc[2 * i + 1] += __log2f(prod2[i].y);
        prod2[i] = (v2f){1.f, 1.f};
      }
    }
    __syncthreads();
  }

  // per-(row, m) log-likelihood into LDS
  const float lw = logw[t];
  #pragma unroll
  for (int r = 0; r < B_TILE; ++r)
    cll[r * M_DIM + t] = fmaf(acc[r], 0.69314718055994530942f, lw);
  __syncthreads();

  // row maxima: 8 waves x 2 rows, shfl reduction
  {
    const int wv = t >> 5, ln = t & 31;
    #pragma unroll
    for (int rr = 0; rr < 2; ++rr) {
      const int row = wv * 2 + rr;
      float m = -INFINITY;
      #pragma unroll
      for (int k = 0; k < M_DIM / 32; ++k)
        m = fmaxf(m, cll[row * M_DIM + k * 32 + ln]);
      #pragma unroll
      for (int off = 16; off; off >>= 1) m = fmaxf(m, __shfl_xor(m, off, 32));
      if (ln == 0) rowmax[row] = m;
    }
  }
  __syncthreads();

  // wave 0: sum_m exp(cll - rowmax) via V_WMMA_F32_16X16X4_F32 with B = ones.
  // A layout (16x4 f32): lanes 0-15 row=lane K={k0,k0+1}; lanes 16-31 K={k0+2,k0+3}
  if (t < 32) {
    const int ln = t;
    const int rr = ln & 15;
    const int kh = (ln >> 4) << 1;
    const float rm = rowmax[rr];
    v8f dacc = {0.f, 0.f, 0.f, 0.f, 0.f, 0.f, 0.f, 0.f};
    const v2f bones = {1.f, 1.f};
    for (int k0 = 0; k0 < M_DIM; k0 += 4) {
      v2f a;
      a.x = __expf(cll[rr * M_DIM + k0 + kh]     - rm);
      a.y = __expf(cll[rr * M_DIM + k0 + kh + 1] - rm);
      dacc = __builtin_amdgcn_wmma_f32_16x16x4_f32(
          false, a, false, bones, (short)0, dacc, false, false);
    }
    // D layout: lane 0 comps 0-7 = rows 0-7; lane 16 comps 0-7 = rows 8-15
    if (ln == 0) {
      #pragma unroll
      for (int v = 0; v < 8; ++v)
        out[b0 + v] = rowmax[v] + __logf(dacc[v]);
    }
    if (ln == 16) {
      #pragma unroll
      for (int v = 0; v < 8; ++v)
        out[b0 + 8 + v] = rowmax[8 + v] + __logf(dacc[v]);
    }
  }
}

// ------------------------------ launcher --------------------------------
extern "C" void kernel_launch(void* const* d_in, const int* in_sizes, int n_in,
                              void* d_out, int out_size, void* d_ws,
                              size_t ws_size, hipStream_t stream) {
  const float* x = (const float*)d_in[0];
  const float* W = (const float*)d_in[1];
  const float* P = (const float*)d_in[2];
  float* out = (float*)d_out;

  float2* ac   = (float2*)d_ws;                                   // 2 MB
  float*  logw = (float*)((char*)d_ws +
                          (size_t)D_DIM * M_DIM * sizeof(float2));

  const int batch = in_sizes[0] / D_DIM;                          // 2048

  prep_logw<<<1, M_DIM, 0, stream>>>(W, logw);
  prep_ac<<<(D_DIM * M_DIM) / 256, 256, 0, stream>>>(P, ac);
  mixll_kernel<<<batch / B_TILE, M_DIM, 0, stream>>>(x, ac, logw, out);
}